// RWRNet_10866267258902
// MI455X (gfx1250) — compile-verified
//
#include <hip/hip_runtime.h>
#include <cstddef>

typedef __attribute__((ext_vector_type(2))) float v2f;
typedef __attribute__((ext_vector_type(8))) float v8f;

#define NNODES 50000
#define DFEAT 128

// ---------------- elementwise / graph helpers ----------------

__global__ void zero_f4(float4* __restrict__ p, int n4) {
  int i = blockIdx.x * blockDim.x + threadIdx.x;
  if (i < n4) p[i] = make_float4(0.f, 0.f, 0.f, 0.f);
}

__global__ void degree_kernel(const int* __restrict__ dst, float* __restrict__ deg, int E) {
  int e = blockIdx.x * blockDim.x + threadIdx.x;
  if (e < E) atomicAdd(&deg[dst[e]], 1.0f);
}

__global__ void rsqrt_kernel(const float* __restrict__ deg, float* __restrict__ dinv, int n) {
  int i = blockIdx.x * blockDim.x + threadIdx.x;
  if (i < n) dinv[i] = rsqrtf(deg[i] + 1.0f);
}

__global__ void coef_kernel(const int* __restrict__ src, const int* __restrict__ dst,
                            const float* __restrict__ dinv, float* __restrict__ coef, int E) {
  int e = blockIdx.x * blockDim.x + threadIdx.x;
  if (e < E) coef[e] = dinv[src[e]] * dinv[dst[e]];
}

// ---------------- fp32 WMMA GEMM ----------------
// C[M,128] = A[M,128] @ W[128,128] (+bias).  FUSE: A := 0.5*A + 0.5*A2 on the fly.
// One block = 8 waves; each wave owns a 16-row x 128-col strip.
// W is pre-swizzled into LDS so each lane's B pair (K=k,k+1) is an aligned
// contiguous b64 at a pure immediate DS offset (no repack movs).
// LDS element for (kg, nt, lane l, j) at  kg*512 + nt*64 + l*2 + j
//   holds W[(4*kg + 2*(l>>4) + j)*128 + nt*16 + (l&15)].

template <bool FUSE>
__global__ __launch_bounds__(256) void gemm128_wmma(
    const float* __restrict__ A, const float* __restrict__ A2,
    const float* __restrict__ W, const float* __restrict__ bias,
    float* __restrict__ C, int M) {
  __shared__ float sB[DFEAT * DFEAT];  // 64 KB, swizzled
  const int t = threadIdx.x;

  // coalesced read of W, swizzled write into LDS
  for (int widx = t; widx < DFEAT * DFEAT; widx += 256) {
    const int k = widx >> 7;        // row of W (K)
    const int n = widx & 127;       // col of W (N)
    const int kg = k >> 2;
    const int half = (k >> 1) & 1;
    const int j = k & 1;
    const int nt = n >> 4;
    const int l = half * 16 + (n & 15);
    sB[kg * 512 + nt * 64 + l * 2 + j] = W[widx];
  }
  __syncthreads();

  const int wave = t >> 5;
  const int lane = t & 31;
  const int half = lane >> 4;
  const int ml   = lane & 15;
  const int m0 = (blockIdx.x * 8 + wave) * 16;
  if (m0 >= M) return;  // wave-uniform; EXEC all-ones in live waves

  v8f acc[8] = {};
  const float* arow  = A + (size_t)(m0 + ml) * DFEAT + 2 * half;
  const float* arow2 = FUSE ? (A2 + (size_t)(m0 + ml) * DFEAT + 2 * half) : nullptr;

  auto loadA = [&](int kg) -> v2f {
    v2f a;
    if (FUSE) {
      a.x = 0.5f * (arow[kg * 4 + 0] + arow2[kg * 4 + 0]);
      a.y = 0.5f * (arow[kg * 4 + 1] + arow2[kg * 4 + 1]);
    } else {
      a.x = arow[kg * 4 + 0];
      a.y = arow[kg * 4 + 1];
    }
    return a;
  };

  const float* bbase0 = sB + lane * 2;
  v2f a = loadA(0);
#pragma unroll 4
  for (int kg = 0; kg < 31; kg++) {
    v2f an = loadA(kg + 1);             // prefetch next A fragment
    const float* bbase = bbase0 + kg * 512;
#pragma unroll
    for (int nt = 0; nt < 8; nt++) {
      v2f b = *(const v2f*)(bbase + nt * 64);
      acc[nt] = __builtin_amdgcn_wmma_f32_16x16x4_f32(
          false, a, false, b, (short)0, acc[nt], false, false);
    }
    a = an;
  }
  {  // last k-group
    const float* bbase = bbase0 + 31 * 512;
#pragma unroll
    for (int nt = 0; nt < 8; nt++) {
      v2f b = *(const v2f*)(bbase + nt * 64);
      acc[nt] = __builtin_amdgcn_wmma_f32_16x16x4_f32(
          false, a, false, b, (short)0, acc[nt], false, false);
    }
  }

  // C/D layout: VGPR v -> row m0 + v + 8*half ; col = nt*16 + ml
#pragma unroll
  for (int nt = 0; nt < 8; nt++) {
    const int col = nt * 16 + ml;
    const float bc = bias ? bias[col] : 0.0f;
#pragma unroll
    for (int v = 0; v < 8; v++) {
      const int row = m0 + v + 8 * half;
      C[(size_t)row * DFEAT + col] = acc[nt][v] + bc;
    }
  }
}

// ---------------- edge scatter: agg[dst] += h[src] * coef ----------------
// Persistent waves, grid-stride over edges; float4 per lane; 4 f32 atomics
// per lane into the (L2-resident) agg matrix.

__global__ void scatter_kernel(const float* __restrict__ h, const int* __restrict__ src,
                               const int* __restrict__ dst, const float* __restrict__ coef,
                               float* __restrict__ agg, int E) {
  const int lane = threadIdx.x & 31;
  const int wid = (blockIdx.x * blockDim.x + threadIdx.x) >> 5;
  const int nwaves = (gridDim.x * blockDim.x) >> 5;
  for (int e = wid; e < E; e += nwaves) {
    const int s = src[e];
    const int d = dst[e];
    const float c = coef[e];
    float4 hv = ((const float4*)(h + (size_t)s * DFEAT))[lane];
    float* out = agg + (size_t)d * DFEAT + lane * 4;
    atomicAdd(out + 0, hv.x * c);
    atomicAdd(out + 1, hv.y * c);
    atomicAdd(out + 2, hv.z * c);
    atomicAdd(out + 3, hv.w * c);
  }
}

// p = relu(agg + h*dinv^2 + bias)  — fused self-loop + bias + activation
__global__ void finish_layer(const float* __restrict__ agg, const float* __restrict__ h,
                             const float* __restrict__ dinv, const float* __restrict__ bias,
                             float* __restrict__ p, int N) {
  int tid = blockIdx.x * blockDim.x + threadIdx.x;
  int row = tid >> 5;
  int lane = tid & 31;
  if (row >= N) return;
  float di = dinv[row];
  float dsq = di * di;
  float4 a  = ((const float4*)(agg + (size_t)row * DFEAT))[lane];
  float4 hv = ((const float4*)(h   + (size_t)row * DFEAT))[lane];
  float4 bv = ((const float4*)bias)[lane];
  float4 r;
  r.x = fmaxf(a.x + hv.x * dsq + bv.x, 0.f);
  r.y = fmaxf(a.y + hv.y * dsq + bv.y, 0.f);
  r.z = fmaxf(a.z + hv.z * dsq + bv.z, 0.f);
  r.w = fmaxf(a.w + hv.w * dsq + bv.w, 0.f);
  ((float4*)(p + (size_t)row * DFEAT))[lane] = r;
}

// out = l2_normalize(0.5*p + 0.5*init) — one wave per node, shuffle reduction
__global__ void final_norm(const float* __restrict__ p, const float* __restrict__ init,
                           float* __restrict__ out, int N) {
  int tid = blockIdx.x * blockDim.x + threadIdx.x;
  int row = tid >> 5;
  int lane = tid & 31;
  if (row >= N) return;
  float4 pv = ((const float4*)(p    + (size_t)row * DFEAT))[lane];
  float4 iv = ((const float4*)(init + (size_t)row * DFEAT))[lane];
  float4 y;
  y.x = 0.5f * (pv.x + iv.x);
  y.y = 0.5f * (pv.y + iv.y);
  y.z = 0.5f * (pv.z + iv.z);
  y.w = 0.5f * (pv.w + iv.w);
  float s = y.x * y.x + y.y * y.y + y.z * y.z + y.w * y.w;
#pragma unroll
  for (int off = 16; off >= 1; off >>= 1) s += __shfl_xor(s, off, 32);
  float scale = 1.0f / fmaxf(sqrtf(s), 1e-12f);
  y.x *= scale; y.y *= scale; y.z *= scale; y.w *= scale;
  ((float4*)(out + (size_t)row * DFEAT))[lane] = y;
}

// ---------------- launch ----------------

extern "C" void kernel_launch(void* const* d_in, const int* in_sizes, int n_in,
                              void* d_out, int out_size, void* d_ws, size_t ws_size,
                              hipStream_t stream) {
  const float* x1    = (const float*)d_in[0];
  const float* x2    = (const float*)d_in[1];
  const int*   ei1   = (const int*)d_in[2];
  const int*   ei2   = (const int*)d_in[3];
  const float* lin_W = (const float*)d_in[4];
  const float* lin_b = (const float*)d_in[5];
  const float* gin_W = (const float*)d_in[6];
  const float* gin_b = (const float*)d_in[7];
  const float* g_W   = (const float*)d_in[8];
  const float* g_b   = (const float*)d_in[9];

  const int N = NNODES;
  const int E = in_sizes[2] / 2;

  // workspace carve-up (floats)
  float* ws   = (float*)d_ws;
  float* init = ws;                          // N*128
  float* p    = init + (size_t)N * DFEAT;    // N*128
  float* t1   = p    + (size_t)N * DFEAT;    // N*128 (agg)
  float* t2   = t1   + (size_t)N * DFEAT;    // N*128 (h)
  float* deg  = t2   + (size_t)N * DFEAT;    // N
  float* dinv = deg + N;                     // N
  float* coef = dinv + N;                    // E

  const int nd4      = N * DFEAT / 4;        // float4 count of a feature matrix
  const int gemmGrid = (N + 127) / 128;      // 8 waves * 16 rows per block
  const int edgeGrid = (E + 255) / 256;
  const int scatGrid = 8192;                 // persistent waves, grid-stride
  const int rowGrid  = (N * 32 + 255) / 256; // one wave per node
  const int zGrid    = (nd4 + 255) / 256;
  const int nGrid    = (N + 255) / 256;

  for (int g = 0; g < 2; g++) {
    const float* x   = g ? x2 : x1;
    const int*   src = g ? ei2 : ei1;
    const int*   dst = src + E;
    float* out = (float*)d_out + (size_t)g * N * DFEAT;

    // normalization coefficients (once per graph, reused by all 3 layers)
    zero_f4<<<(N / 4 + 255) / 256, 256, 0, stream>>>((float4*)deg, N / 4);
    degree_kernel<<<edgeGrid, 256, 0, stream>>>(dst, deg, E);
    rsqrt_kernel<<<nGrid, 256, 0, stream>>>(deg, dinv, N);
    coef_kernel<<<edgeGrid, 256, 0, stream>>>(src, dst, dinv, coef, E);

    // init = x @ lin_W + lin_b
    gemm128_wmma<false><<<gemmGrid, 256, 0, stream>>>(x, nullptr, lin_W, lin_b, init, N);

    // layer 1: h = x @ gcn_in_W ; agg ; p = relu(agg + h*dinv^2 + b)
    gemm128_wmma<false><<<gemmGrid, 256, 0, stream>>>(x, nullptr, gin_W, nullptr, t2, N);
    zero_f4<<<zGrid, 256, 0, stream>>>((float4*)t1, nd4);
    scatter_kernel<<<scatGrid, 256, 0, stream>>>(t2, src, dst, coef, t1, E);
    finish_layer<<<rowGrid, 256, 0, stream>>>(t1, t2, dinv, gin_b, p, N);

    // layers 2..3 (shared weights): h = (0.5*p + 0.5*init) @ gcn_W, combine fused
    for (int l = 0; l < 2; l++) {
      gemm128_wmma<true><<<gemmGrid, 256, 0, stream>>>(p, init, g_W, nullptr, t2, N);
      zero_f4<<<zGrid, 256, 0, stream>>>((float4*)t1, nd4);
      scatter_kernel<<<scatGrid, 256, 0, stream>>>(t2, src, dst, coef, t1, E);
      finish_layer<<<rowGrid, 256, 0, stream>>>(t1, t2, dinv, g_b, p, N);
    }

    final_norm<<<rowGrid, 256, 0, stream>>>(p, init, out, N);
  }
}